// IntegratedTemporalGraphMemory_83064667505147
// MI455X (gfx1250) — compile-verified
//
#include <hip/hip_runtime.h>
#include <hip/hip_bf16.h>
#include <math.h>

// ---------------------------------------------------------------------------
// Sizes from the reference
//   SD=64 EMO=16 D=80 C=64 E=320 NH=4 HD=80 F=400
//   outputs: causal_adj 729 | attended 8640 | measured 8640 | new_root 400
// ---------------------------------------------------------------------------

typedef float v2f __attribute__((ext_vector_type(2)));
typedef float v8f __attribute__((ext_vector_type(8)));

// ---------------- generic f32 WMMA GEMM: one wave per 16x16 tile ------------
// C[m,n] = relu?( ( sum_k A[m,k]*B[k,n] + bias[n] ) * alpha )
// A row-major (lda).  B accessed as B[k*b_rs + n*b_cs] (supports W and W^T).
//
// Out-of-range M rows / N cols are handled by CLAMPING the index (loads stay
// in-bounds, garbage values only reach C rows/cols that the store guard
// discards) -> no exec-masked loads in the hot loop.  Only the K tail needs a
// real guard, peeled out of the main loop.
__global__ void gemm_wmma(const float* __restrict__ A, int lda,
                          const float* __restrict__ B, int b_rs, int b_cs,
                          float* __restrict__ C, int ldc,
                          int M, int N, int K,
                          float alpha, const float* __restrict__ bias, int relu)
{
    int tilesN = (N + 15) >> 4;
    int t  = blockIdx.x;
    int tm = t / tilesN, tn = t % tilesN;
    int m0 = tm << 4, n0 = tn << 4;

    int lane = threadIdx.x & 31;
    int half = lane >> 4;       // 0: lanes 0-15, 1: lanes 16-31
    int lq   = lane & 15;

    int arow = m0 + lq;
    int bcol = n0 + lq;
    int arow_c = arow < M ? arow : M - 1;   // clamp: garbage rows are discarded
    int bcol_c = bcol < N ? bcol : N - 1;   // clamp: garbage cols are discarded

    const float* Ap = A + arow_c * lda;     // per-lane row base
    const float* Bp = B + bcol_c * b_cs;    // per-lane col base

    v8f acc = {};
    int K4 = K & ~3;
#pragma unroll 2
    for (int k0 = 0; k0 < K4; k0 += 4) {
        int ka = k0 + 2 * half;             // this half's K pair: {0,1} or {2,3}
        v2f a, b;
        a.x = Ap[ka];
        a.y = Ap[ka + 1];
        b.x = Bp[ka * b_rs];
        b.y = Bp[(ka + 1) * b_rs];
        acc = __builtin_amdgcn_wmma_f32_16x16x4_f32(
                  false, a, false, b, (short)0, acc, false, false);
    }
    if (K4 < K) {                           // K tail (e.g. K=27 attention dims)
        int ka = K4 + 2 * half;
        v2f a = {0.f, 0.f}, b = {0.f, 0.f};
        if (ka < K)     { a.x = Ap[ka];     b.x = Bp[ka * b_rs]; }
        if (ka + 1 < K) { a.y = Ap[ka + 1]; b.y = Bp[(ka + 1) * b_rs]; }
        acc = __builtin_amdgcn_wmma_f32_16x16x4_f32(
                  false, a, false, b, (short)0, acc, false, false);
    }

    if (bcol >= N) return;
    float bv = bias ? bias[bcol] : 0.f;
#pragma unroll
    for (int r = 0; r < 8; ++r) {
        int orow = m0 + r + 8 * half;       // C VGPR r: M=r (lo half), M=8+r (hi half)
        if (orow < M) {
            float v = (acc[r] + bv) * alpha;
            if (relu) v = fmaxf(v, 0.f);
            C[orow * ldc + bcol] = v;
        }
    }
}

// ---------------- causal adjacency ------------------------------------------
__global__ void causal_k(const float* __restrict__ node, const float* __restrict__ hist,
                         const float* __restrict__ gW, const float* __restrict__ gb,
                         float* __restrict__ adj)
{
    __shared__ float av[27], bv[27];
    int tid = threadIdx.x;
    if (tid < 27) {
        float a = 0.f, b = 0.f;
        for (int d = 0; d < 320; ++d) {
            a += hist[tid * 320 + d] * gW[d];
            b += node[tid * 320 + d] * gW[320 + d];
        }
        av[tid] = a; bv[tid] = b;
    }
    __syncthreads();
    float bias = gb[0];
    for (int e = tid; e < 729; e += blockDim.x) {
        int i = e / 27, j = e - i * 27;
        adj[e] = (i == j) ? 0.f : 1.f / (1.f + __expf(-(av[j] + bv[i] + bias)));
    }
}

// ---------------- softmax over 27 keys, one wave per (head,query) row -------
__global__ void softmax27(float* __restrict__ attn)
{
    int r = blockIdx.x;                  // 0..107
    int h = r / 27, q = r - h * 27;
    float* row = attn + h * 729 + q * 27;
    int lane = threadIdx.x;
    float x = (lane < 27) ? row[lane] : -3.4e38f;
    float m = x;
    for (int o = 16; o; o >>= 1) m = fmaxf(m, __shfl_xor(m, o, 32));
    float e = (lane < 27) ? __expf(x - m) : 0.f;
    float s = e;
    for (int o = 16; o; o >>= 1) s += __shfl_xor(s, o, 32);
    if (lane < 27) row[lane] = e / s;
}

// ---------------- quantum: build 351x128 pair matrix ------------------------
__device__ __forceinline__ void pair_from_p(int p, int& i, int& j)
{
    i = 0; int rem = p;
    while (rem >= 26 - i) { rem -= 26 - i; ++i; }
    j = i + 1 + rem;
}

__global__ void build_comb(const float* __restrict__ q_real, float* __restrict__ comb)
{
    int idx = blockIdx.x * blockDim.x + threadIdx.x;
    if (idx >= 351 * 128) return;
    int p = idx >> 7, c = idx & 127;
    int i, j; pair_from_p(p, i, j);
    comb[idx] = (c < 64) ? q_real[i * 64 + c] : q_real[j * 64 + (c - 64)];
}

// ---------------- quantum: coupling, averaged entanglement, qr/qi -----------
__global__ void quantum_mid(const float* __restrict__ q_real, const float* __restrict__ q_imag,
                            const float* __restrict__ ent,
                            float* __restrict__ avg_out, float* __restrict__ qrqi)
{
    __shared__ float s[27];
    __shared__ float coup[351];
    __shared__ float avg[64];
    int tid = threadIdx.x;               // 128 threads
    if (tid < 27) {
        float a = 0.f;
        for (int c = 0; c < 64; ++c) a += q_real[tid * 64 + c];
        s[tid] = a;
    }
    __syncthreads();
    for (int p = tid; p < 351; p += 128) {
        int i, j; pair_from_p(p, i, j);
        coup[p] = __expf(-fabsf(s[i] - s[j]) / (500.0f + 1e-6f));
    }
    __syncthreads();
    if (tid < 64) {
        float a = 0.f;
        for (int p = 0; p < 351; ++p) a += ent[p * 64 + tid] * coup[p];
        avg[tid] = a / 351.0f;
        avg_out[tid] = avg[tid];
    }
    __syncthreads();
    for (int idx = tid; idx < 27 * 128; idx += 128) {
        int r = idx >> 7, c2 = idx & 127;
        float base = (c2 < 64) ? q_real[r * 64 + c2] : q_imag[r * 64 + (c2 - 64)];
        qrqi[idx] = base + avg[c2 & 63];
    }
}

// ---------------- fractal: init agg buffers with biases ---------------------
__global__ void init_fract(const float* __restrict__ mid_agg_b,
                           const float* __restrict__ root_agg_b,
                           float* __restrict__ agg, float* __restrict__ aggr)
{
    int idx = blockIdx.x * blockDim.x + threadIdx.x;
    if (idx < 10800)       agg[idx] = mid_agg_b[idx];
    else if (idx < 11200)  aggr[idx - 10800] = root_agg_b[idx - 10800];
}

__device__ __forceinline__ void atomic_add_f32(float* p, float v)
{
    __hip_atomic_fetch_add(p, v, __ATOMIC_RELAXED, __HIP_MEMORY_SCOPE_AGENT);
}

// ---------------- fractal: big streaming matvec (467 MB, HBM-bound) ---------
// agg[n,j] += sum_{i in chunk} leaf[n,i] * W[n,i,j]
#define MCHUNK 100
__global__ void matvec_mid(const float* __restrict__ x,   // leaf, 27*10800
                           const float* __restrict__ W,   // 27*10800*400
                           float* __restrict__ agg)       // 27*400 (bias pre-init)
{
    __shared__ float xs[MCHUNK];
    int n  = blockIdx.x;
    int i0 = blockIdx.y * MCHUNK;
    int tid = threadIdx.x;
    if (tid < MCHUNK) xs[tid] = x[n * 10800 + i0 + tid];
    __syncthreads();
    if (tid >= 400) return;
    const float* Wb = W + ((size_t)n * 10800 + i0) * 400 + tid;
    float acc = 0.f;
#pragma unroll 4
    for (int ii = 0; ii < MCHUNK; ++ii) acc += xs[ii] * Wb[(size_t)ii * 400];
    atomic_add_f32(&agg[n * 400 + tid], acc);
}

__global__ void matvec_root(const float* __restrict__ cf,  // 10800
                            const float* __restrict__ W,   // 10800*400
                            float* __restrict__ aggr)      // 400 (bias pre-init)
{
    __shared__ float xs[MCHUNK];
    int i0 = blockIdx.x * MCHUNK;
    int tid = threadIdx.x;
    if (tid < MCHUNK) xs[tid] = cf[i0 + tid];
    __syncthreads();
    if (tid >= 400) return;
    const float* Wb = W + (size_t)i0 * 400 + tid;
    float acc = 0.f;
#pragma unroll 4
    for (int ii = 0; ii < MCHUNK; ++ii) acc += xs[ii] * Wb[(size_t)ii * 400];
    atomic_add_f32(&aggr[tid], acc);
}

// ---------------- fractal: per-node update MLP ------------------------------
__global__ void mid_mlp(const float* __restrict__ mid, const float* __restrict__ agg,
                        const float* __restrict__ W1, const float* __restrict__ b1,
                        const float* __restrict__ W2, const float* __restrict__ b2,
                        float* __restrict__ cf)
{
    __shared__ float comb[800];
    __shared__ float hid[64];
    int n = blockIdx.x, tid = threadIdx.x;
    for (int idx = tid; idx < 800; idx += blockDim.x)
        comb[idx] = (idx < 400) ? mid[n * 400 + idx] : agg[n * 400 + idx - 400];
    __syncthreads();
    if (tid < 64) {
        float acc = b1[n * 64 + tid];
        const float* w = W1 + (size_t)n * 800 * 64 + tid;
        for (int i = 0; i < 800; ++i) acc += comb[i] * w[(size_t)i * 64];
        hid[tid] = fmaxf(acc, 0.f);
    }
    __syncthreads();
    if (tid < 400) {
        float acc = b2[n * 400 + tid];
        const float* w = W2 + (size_t)n * 64 * 400 + tid;
#pragma unroll
        for (int h = 0; h < 64; ++h) acc += hid[h] * w[(size_t)h * 400];
        cf[n * 400 + tid] = mid[n * 400 + tid] + acc;
    }
}

__global__ void root_mlp(const float* __restrict__ root, const float* __restrict__ aggr,
                         const float* __restrict__ W1, const float* __restrict__ b1,
                         const float* __restrict__ W2, const float* __restrict__ b2,
                         float* __restrict__ out)
{
    __shared__ float comb[800];
    __shared__ float hid[64];
    int tid = threadIdx.x;
    for (int idx = tid; idx < 800; idx += blockDim.x)
        comb[idx] = (idx < 400) ? root[idx] : aggr[idx - 400];
    __syncthreads();
    if (tid < 64) {
        float acc = b1[tid];
        for (int i = 0; i < 800; ++i) acc += comb[i] * W1[i * 64 + tid];
        hid[tid] = fmaxf(acc, 0.f);
    }
    __syncthreads();
    if (tid < 400) {
        float acc = b2[tid];
#pragma unroll
        for (int h = 0; h < 64; ++h) acc += hid[h] * W2[h * 400 + tid];
        out[tid] = root[tid] + acc;
    }
}

// ---------------------------------------------------------------------------
// Workspace layout (floats)
// ---------------------------------------------------------------------------
enum {
    WS_QKV  = 0,                 // 27*960   = 25920
    WS_ATTN = WS_QKV  + 25920,   // 4*27*27  =  2916
    WS_O    = WS_ATTN +  2916,   // 27*320   =  8640
    WS_COMB = WS_O    +  8640,   // 351*128  = 44928
    WS_ENTH = WS_COMB + 44928,   // 351*64   = 22464
    WS_ENT  = WS_ENTH + 22464,   // 351*64   = 22464
    WS_AVG  = WS_ENT  + 22464,   // 64
    WS_QRQI = WS_AVG  +    64,   // 27*128   =  3456
    WS_AGG  = WS_QRQI +  3456,   // 27*400   = 10800
    WS_CF   = WS_AGG  + 10800,   // 27*400   = 10800
    WS_AGGR = WS_CF   + 10800,   // 400
};

extern "C" void kernel_launch(void* const* d_in, const int* in_sizes, int n_in,
                              void* d_out, int out_size, void* d_ws, size_t ws_size,
                              hipStream_t stream)
{
    const float* node       = (const float*)d_in[0];
    const float* hist       = (const float*)d_in[1];
    const float* gW         = (const float*)d_in[2];
    const float* gb         = (const float*)d_in[3];
    const float* mha_in_W   = (const float*)d_in[4];
    const float* mha_in_b   = (const float*)d_in[5];
    const float* mha_out_W  = (const float*)d_in[6];
    const float* mha_out_b  = (const float*)d_in[7];
    const float* q_real     = (const float*)d_in[8];
    const float* q_imag     = (const float*)d_in[9];
    const float* ent_W1     = (const float*)d_in[10];
    const float* ent_b1     = (const float*)d_in[11];
    const float* ent_W2     = (const float*)d_in[12];
    const float* ent_b2     = (const float*)d_in[13];
    const float* meas_W     = (const float*)d_in[14];
    const float* meas_b     = (const float*)d_in[15];
    const float* leaf       = (const float*)d_in[16];
    const float* mid        = (const float*)d_in[17];
    const float* root       = (const float*)d_in[18];
    const float* mid_agg_W  = (const float*)d_in[19];
    const float* mid_agg_b  = (const float*)d_in[20];
    const float* root_agg_W = (const float*)d_in[21];
    const float* root_agg_b = (const float*)d_in[22];
    const float* mid_up_W1  = (const float*)d_in[23];
    const float* mid_up_b1  = (const float*)d_in[24];
    const float* mid_up_W2  = (const float*)d_in[25];
    const float* mid_up_b2  = (const float*)d_in[26];
    const float* root_up_W1 = (const float*)d_in[27];
    const float* root_up_b1 = (const float*)d_in[28];
    const float* root_up_W2 = (const float*)d_in[29];
    const float* root_up_b2 = (const float*)d_in[30];

    float* out      = (float*)d_out;
    float* ws       = (float*)d_ws;
    float* adj_out  = out;           // 729
    float* att_out  = out + 729;     // 8640
    float* mea_out  = out + 9369;    // 8640
    float* root_out = out + 18009;   // 400

    auto gemm = [&](const float* A, int lda, const float* B, int brs, int bcs,
                    float* C, int ldc, int M, int N, int K,
                    float alpha, const float* bias, int relu) {
        int tiles = ((M + 15) / 16) * ((N + 15) / 16);
        gemm_wmma<<<tiles, 32, 0, stream>>>(A, lda, B, brs, bcs, C, ldc,
                                            M, N, K, alpha, bias, relu);
    };

    // ---- fractal agg init (bias) runs first --------------------------------
    init_fract<<<(11200 + 511) / 512, 512, 0, stream>>>(mid_agg_b, root_agg_b,
                                                        ws + WS_AGG, ws + WS_AGGR);

    // ---- causal adjacency --------------------------------------------------
    causal_k<<<1, 512, 0, stream>>>(node, hist, gW, gb, adj_out);

    // ---- MHA ---------------------------------------------------------------
    // qkv = x @ W_in^T + b_in   (27x960), W_in is (960,320) row-major -> B=W^T
    gemm(node, 320, mha_in_W, 1, 320, ws + WS_QKV, 960, 27, 960, 320, 1.f, mha_in_b, 0);
    const float scale = 0.11180339887498948f;   // 1/sqrt(80)
    for (int h = 0; h < 4; ++h)                 // logits = Q K^T / sqrt(HD)
        gemm(ws + WS_QKV + h * 80, 960,
             ws + WS_QKV + 320 + h * 80, 1, 960,
             ws + WS_ATTN + h * 729, 27, 27, 27, 80, scale, nullptr, 0);
    softmax27<<<108, 32, 0, stream>>>(ws + WS_ATTN);
    for (int h = 0; h < 4; ++h)                 // o = attn @ V
        gemm(ws + WS_ATTN + h * 729, 27,
             ws + WS_QKV + 640 + h * 80, 960, 1,
             ws + WS_O + h * 80, 320, 27, 80, 27, 1.f, nullptr, 0);
    // attended = (o @ W_out^T + b_out) * hertz_w(=0.5)
    gemm(ws + WS_O, 320, mha_out_W, 1, 320, att_out, 320, 27, 320, 320, 0.5f, mha_out_b, 0);

    // ---- Quantum-entangled GNN --------------------------------------------
    build_comb<<<(351 * 128 + 511) / 512, 512, 0, stream>>>(q_real, ws + WS_COMB);
    gemm(ws + WS_COMB, 128, ent_W1, 64, 1, ws + WS_ENTH, 64, 351, 64, 128, 1.f, ent_b1, 1);
    gemm(ws + WS_ENTH, 64,  ent_W2, 64, 1, ws + WS_ENT,  64, 351, 64, 64,  1.f, ent_b2, 0);
    quantum_mid<<<1, 128, 0, stream>>>(q_real, q_imag, ws + WS_ENT,
                                       ws + WS_AVG, ws + WS_QRQI);
    gemm(ws + WS_QRQI, 128, meas_W, 320, 1, mea_out, 320, 27, 320, 128, 1.f, meas_b, 0);

    // ---- FractalGNN --------------------------------------------------------
    matvec_mid<<<dim3(27, 108), 512, 0, stream>>>(leaf, mid_agg_W, ws + WS_AGG);
    mid_mlp<<<27, 512, 0, stream>>>(mid, ws + WS_AGG, mid_up_W1, mid_up_b1,
                                    mid_up_W2, mid_up_b2, ws + WS_CF);
    matvec_root<<<108, 512, 0, stream>>>(ws + WS_CF, root_agg_W, ws + WS_AGGR);
    root_mlp<<<1, 512, 0, stream>>>(root, ws + WS_AGGR, root_up_W1, root_up_b1,
                                    root_up_W2, root_up_b2, root_out);
}